// BatchHardTripletLoss_39977555591697
// MI455X (gfx1250) — compile-verified
//
#include <hip/hip_runtime.h>

typedef __attribute__((ext_vector_type(2))) float v2f;
typedef __attribute__((ext_vector_type(4))) float v4f;
typedef __attribute__((ext_vector_type(8))) float v8f;

#define N_ROWS   2048
#define N2       4096
#define D        512
#define TM       64          // A rows per block
#define TN       128         // B rows per block
#define KC       32
#define LDSD     36          // padded LDS row stride (floats): conflict-free frag reads
#define ALPHA    0.1f
#define REL_T    1e-5f
#define INF_BITS 0x7F800000u

__device__ __forceinline__ const float* batch_row(const float* h1, const float* h2, int i) {
  return (i < N_ROWS) ? (h1 + (size_t)i * D) : (h2 + (size_t)(i - N_ROWS) * D);
}

// ---------------- squared norms + init per-row neg-min --------------------
__global__ __launch_bounds__(128)
void sq_init_kernel(const float* __restrict__ h1, const float* __restrict__ h2,
                    float* __restrict__ sq, unsigned* __restrict__ negmin) {
  const int i = blockIdx.x;
  const v4f* r4 = (const v4f*)batch_row(h1, h2, i);
  v4f x = r4[threadIdx.x];                       // 128 threads * float4 = 512 floats
  float s = x.x * x.x + x.y * x.y + x.z * x.z + x.w * x.w;
  for (int off = 16; off; off >>= 1) s += __shfl_xor(s, off, 32);
  __shared__ float part[4];
  if ((threadIdx.x & 31) == 0) part[threadIdx.x >> 5] = s;
  __syncthreads();
  if (threadIdx.x == 0) {
    sq[i] = part[0] + part[1] + part[2] + part[3];
    negmin[i] = INF_BITS;
  }
}

// ---------------- the single positive distance per row --------------------
__global__ __launch_bounds__(128)
void pos_kernel(const float* __restrict__ h1, const float* __restrict__ h2,
                const float* __restrict__ sq, float* __restrict__ posd) {
  const int i = blockIdx.x;
  const v4f* a4 = (const v4f*)(h1 + (size_t)i * D);
  const v4f* b4 = (const v4f*)(h2 + (size_t)i * D);
  v4f a = a4[threadIdx.x];
  v4f b = b4[threadIdx.x];
  float s = a.x * b.x + a.y * b.y + a.z * b.z + a.w * b.w;
  for (int off = 16; off; off >>= 1) s += __shfl_xor(s, off, 32);
  __shared__ float part[4];
  if ((threadIdx.x & 31) == 0) part[threadIdx.x >> 5] = s;
  __syncthreads();
  if (threadIdx.x == 0) {
    float dot = part[0] + part[1] + part[2] + part[3];
    float d2 = sq[i] + sq[i + N_ROWS] - 2.0f * dot;
    posd[i] = fmaxf(sqrtf(fmaxf(d2, 1e-14f)), 1e-7f);
  }
}

// ------- fused WMMA gram tile + distance + hardest-negative min -----------
// Block tile: 64 (M) x 128 (N).  8 waves, each owning a 2x2 grid of 16x16
// WMMA tiles -> 4 accumulators, 4 fragment loads per 4 WMMAs.
__global__ __launch_bounds__(256)
void gram_min_kernel(const float* __restrict__ h1, const float* __restrict__ h2,
                     const float* __restrict__ sq, unsigned* __restrict__ negmin) {
  __shared__ float As[TM * LDSD];       //  64 x 36 floats =  9 KB
  __shared__ float Bs[TN * LDSD];       // 128 x 36 floats = 18 KB

  const int t    = threadIdx.x;
  const int lane = t & 31;
  const int wave = t >> 5;
  const int r    = lane & 15;           // row/col within 16-wide tile
  const int half = lane >> 4;           // K-half selector for fragments
  const int m0   = (wave & 1) * 2;      // M-subtiles {m0, m0+1} of 4
  const int m1   = m0 + 1;
  const int n0   = (wave >> 1) * 2;     // N-subtiles {n0, n0+1} of 8
  const int n1   = n0 + 1;
  const int rowBase = blockIdx.x * TM;
  const int colBase = blockIdx.y * TN;

  v8f acc00 = {}, acc01 = {}, acc10 = {}, acc11 = {};

  for (int kc = 0; kc < D; kc += KC) {
    __syncthreads();
    // stage A: 64x32 = 512 float4 (2/thread); B: 128x32 = 1024 float4 (4/thread)
    #pragma unroll
    for (int i = 0; i < 2; ++i) {
      int f4  = t + i * 256;
      int row = f4 >> 3;                // 8 float4 per 32-float row
      int c4  = f4 & 7;
      v4f va = *(const v4f*)(batch_row(h1, h2, rowBase + row) + kc + c4 * 4);
      *(v4f*)&As[row * LDSD + c4 * 4] = va;
    }
    #pragma unroll
    for (int i = 0; i < 4; ++i) {
      int f4  = t + i * 256;
      int row = f4 >> 3;
      int c4  = f4 & 7;
      v4f vb = *(const v4f*)(batch_row(h1, h2, colBase + row) + kc + c4 * 4);
      *(v4f*)&Bs[row * LDSD + c4 * 4] = vb;
    }
    __syncthreads();

    if (kc + KC < D) {                  // near-cache prefetch of next K chunk
      int rowA = t & 63;
      __builtin_prefetch(batch_row(h1, h2, rowBase + rowA) + kc + KC, 0, 3);
      int rowB = t & 127;
      __builtin_prefetch(batch_row(h1, h2, colBase + rowB) + kc + KC, 0, 3);
    }

    #pragma unroll
    for (int k = 0; k < KC; k += 4) {
      const int ka = k + 2 * half;      // K pair for this lane-half
      v2f af0 = *(const v2f*)&As[(m0 * 16 + r) * LDSD + ka];
      v2f af1 = *(const v2f*)&As[(m1 * 16 + r) * LDSD + ka];
      v2f bf0 = *(const v2f*)&Bs[(n0 * 16 + r) * LDSD + ka];
      v2f bf1 = *(const v2f*)&Bs[(n1 * 16 + r) * LDSD + ka];
      acc00 = __builtin_amdgcn_wmma_f32_16x16x4_f32(false, af0, false, bf0,
                                                    (short)0, acc00, false, false);
      acc01 = __builtin_amdgcn_wmma_f32_16x16x4_f32(false, af0, false, bf1,
                                                    (short)0, acc01, false, false);
      acc10 = __builtin_amdgcn_wmma_f32_16x16x4_f32(false, af1, false, bf0,
                                                    (short)0, acc10, false, false);
      acc11 = __builtin_amdgcn_wmma_f32_16x16x4_f32(false, af1, false, bf1,
                                                    (short)0, acc11, false, false);
    }
  }

  // epilogue: gram -> distance -> mask self/positive -> per-row min -> atomic
  const int   gj0  = colBase + n0 * 16 + r;
  const int   gj1  = colBase + n1 * 16 + r;
  const float sqj0 = sq[gj0];
  const float sqj1 = sq[gj1];

  auto reduce_rows = [&](const v8f& a0, const v8f& a1, int m) {
    #pragma unroll
    for (int v = 0; v < 8; ++v) {
      int gi = rowBase + m * 16 + half * 8 + v;    // C/D layout: M = v + 8*half
      float sqi = sq[gi];
      float d0 = fmaxf(sqrtf(fmaxf(sqi + sqj0 - 2.0f * a0[v], 1e-14f)), 1e-7f);
      float d1 = fmaxf(sqrtf(fmaxf(sqi + sqj1 - 2.0f * a1[v], 1e-14f)), 1e-7f);
      int partner = gi ^ N_ROWS;                   // (gi + 2048) mod 4096
      if (gj0 == gi || gj0 == partner) d0 = __builtin_inff();
      if (gj1 == gi || gj1 == partner) d1 = __builtin_inff();
      float mn = fminf(d0, d1);                    // same row gi across both N tiles
      for (int off = 8; off; off >>= 1) mn = fminf(mn, __shfl_xor(mn, off, 32));
      if (r == 0)                                  // lanes 0 and 16: distinct rows
        atomicMin(&negmin[gi], __float_as_uint(mn)); // dist>=0: uint order==float order
    }
  };
  reduce_rows(acc00, acc01, m0);
  reduce_rows(acc10, acc11, m1);
}

// ---------------- final 5-scalar reduction --------------------------------
__global__ __launch_bounds__(256)
void finalize_kernel(const float* __restrict__ sq, const unsigned* __restrict__ negmin,
                     const float* __restrict__ posd, float* __restrict__ out) {
  float sdiff = 0.f, srel = 0.f, ssq = 0.f;
  int good = 0, nrel = 0;
  for (int i = threadIdx.x; i < N2; i += 256) {
    float neg  = __uint_as_float(negmin[i]);
    float pos  = posd[i & (N_ROWS - 1)];
    float diff = pos - neg;
    float tl   = fmaxf(diff + ALPHA, 0.f);
    sdiff += diff;
    ssq   += sq[i];
    if (tl < REL_T) good++;
    if (tl > REL_T) { srel += tl; nrel++; }
  }
  for (int off = 16; off; off >>= 1) {
    sdiff += __shfl_xor(sdiff, off, 32);
    srel  += __shfl_xor(srel,  off, 32);
    ssq   += __shfl_xor(ssq,   off, 32);
    good  += __shfl_xor(good,  off, 32);
    nrel  += __shfl_xor(nrel,  off, 32);
  }
  __shared__ float fs[8][3];
  __shared__ int   is[8][2];
  int wid = threadIdx.x >> 5;
  if ((threadIdx.x & 31) == 0) {
    fs[wid][0] = sdiff; fs[wid][1] = srel; fs[wid][2] = ssq;
    is[wid][0] = good;  is[wid][1] = nrel;
  }
  __syncthreads();
  if (threadIdx.x == 0) {
    float a = 0.f, b = 0.f, c = 0.f; int g = 0, nr = 0;
    for (int w = 0; w < 8; ++w) {
      a += fs[w][0]; b += fs[w][1]; c += fs[w][2];
      g += is[w][0]; nr += is[w][1];
    }
    int nrc = nr > 0 ? nr : 1;
    out[0] = b / (float)nrc;          // loss (mean of relevant triplets)
    out[1] = a / (float)N2;           // mean(differences)
    out[2] = (float)g;                // good
    out[3] = (float)(N2 - g);         // bad
    out[4] = sqrtf(c / (float)N2);    // sqrt(mean ||x||^2)
  }
}

extern "C" void kernel_launch(void* const* d_in, const int* in_sizes, int n_in,
                              void* d_out, int out_size, void* d_ws, size_t ws_size,
                              hipStream_t stream) {
  const float* h1 = (const float*)d_in[0];
  const float* h2 = (const float*)d_in[1];
  // h3 (d_in[2]) is unused by the reference.

  float*    sq     = (float*)d_ws;                                    // 4096 f32
  unsigned* negmin = (unsigned*)((char*)d_ws + N2 * sizeof(float));   // 4096 u32
  float*    posd   = (float*)((char*)d_ws + 2 * N2 * sizeof(float));  // 2048 f32

  sq_init_kernel<<<N2, 128, 0, stream>>>(h1, h2, sq, negmin);
  pos_kernel<<<N_ROWS, 128, 0, stream>>>(h1, h2, sq, posd);
  dim3 grid(N2 / TM, N2 / TN);
  gram_min_kernel<<<grid, 256, 0, stream>>>(h1, h2, sq, negmin);
  finalize_kernel<<<1, 256, 0, stream>>>(sq, negmin, posd, (float*)d_out);
}